// BilinearPooling_36490042147537
// MI455X (gfx1250) — compile-verified
//
#include <hip/hip_runtime.h>

// BilinearPooling collapsed: out[i,k] = rowmean(c1)[i] * c2[i,k]
// B = 256 rows, J = K = 14*14*256 = 50176 contiguous fp32 per row.
//
// Memory-bound: ~154 MB total traffic -> ~6.6 us at 23.3 TB/s.
// Row reduction uses V_WMMA_F32_16X16X4_F32 with A == ones: each WMMA
// accumulates 64 fp32 wave values into D. Sum of ALL 256 D entries equals
// 16 * (sum of all B inputs) independent of the exact lane<->(k,n) mapping,
// so the reduction is correct for any bijective VGPR layout.

typedef __attribute__((ext_vector_type(2))) float v2f;
typedef __attribute__((ext_vector_type(8))) float v8f;

#define ROWS        256
#define COLS        50176      // 14*14*256
#define CHUNKS      7          // chunks per row
#define CHUNK_ELEMS 7168       // COLS / CHUNKS
#define TPB         256        // 8 waves (wave32)
#define ITERS       7          // CHUNK_ELEMS / (TPB * 4)
#define NBLOCKS     (ROWS * CHUNKS)   // 1792

__global__ __launch_bounds__(TPB)
void bp_reduce_rows(const float* __restrict__ c1, float* __restrict__ partials) {
    const int row   = blockIdx.x / CHUNKS;
    const int chunk = blockIdx.x % CHUNKS;
    const float4* __restrict__ src =
        (const float4*)(c1 + (size_t)row * COLS + (size_t)chunk * CHUNK_ELEMS);
    const int tid = threadIdx.x;

    v8f acc = {0.f, 0.f, 0.f, 0.f, 0.f, 0.f, 0.f, 0.f};
    const v2f ones = {1.0f, 1.0f};

    // Hot loop: 1x b128 load + 2x v_wmma_f32_16x16x4_f32 per iteration.
    // EXEC is all-ones for every wave here (no divergence before this point).
#pragma unroll
    for (int it = 0; it < ITERS; ++it) {
        float4 v = src[it * TPB + tid];
        v2f b0; b0.x = v.x; b0.y = v.y;
        v2f b1; b1.x = v.z; b1.y = v.w;
        acc = __builtin_amdgcn_wmma_f32_16x16x4_f32(
            false, ones, false, b0, (short)0, acc, false, false);
        acc = __builtin_amdgcn_wmma_f32_16x16x4_f32(
            false, ones, false, b1, (short)0, acc, false, false);
    }

    // Sum this lane's 8 D entries; 32 lanes together cover all 256 D entries.
    float s = acc[0] + acc[1] + acc[2] + acc[3] + acc[4] + acc[5] + acc[6] + acc[7];
#pragma unroll
    for (int off = 16; off >= 1; off >>= 1)
        s += __shfl_xor(s, off, 32);
    s *= (1.0f / 16.0f);   // each B column-sum is replicated across 16 D rows

    // Combine the 8 waves of this block (deterministic fixed order).
    __shared__ float ws[TPB / 32];
    const int wave = tid >> 5;
    const int lane = tid & 31;
    if (lane == 0) ws[wave] = s;
    __syncthreads();
    if (tid == 0) {
        float t = 0.f;
#pragma unroll
        for (int w = 0; w < TPB / 32; ++w) t += ws[w];
        partials[blockIdx.x] = t;   // partial sum of this (row, chunk)
    }
}

__global__ __launch_bounds__(TPB)
void bp_scale_rows(const float* __restrict__ c2,
                   const float* __restrict__ partials,
                   float* __restrict__ out) {
    const int row   = blockIdx.x / CHUNKS;
    const int chunk = blockIdx.x % CHUNKS;

    // Uniform (wave-scalar) re-sum of the 7 row partials, fixed order.
    float m = 0.f;
#pragma unroll
    for (int i = 0; i < CHUNKS; ++i) m += partials[row * CHUNKS + i];
    m *= (1.0f / (float)COLS);   // row mean

    const size_t off = (size_t)row * COLS + (size_t)chunk * CHUNK_ELEMS;
    const float4* __restrict__ src = (const float4*)(c2 + off);
    float4*       __restrict__ dst = (float4*)(out + off);
    const int tid = threadIdx.x;

#pragma unroll
    for (int it = 0; it < ITERS; ++it) {
        float4 v = src[it * TPB + tid];
        v.x *= m; v.y *= m; v.z *= m; v.w *= m;
        dst[it * TPB + tid] = v;
    }
}

extern "C" void kernel_launch(void* const* d_in, const int* in_sizes, int n_in,
                              void* d_out, int out_size, void* d_ws, size_t ws_size,
                              hipStream_t stream) {
    const float* c1 = (const float*)d_in[0];
    const float* c2 = (const float*)d_in[1];
    float* out      = (float*)d_out;
    float* partials = (float*)d_ws;   // NBLOCKS floats = 7168 B, fully overwritten

    bp_reduce_rows<<<NBLOCKS, TPB, 0, stream>>>(c1, partials);
    bp_scale_rows<<<NBLOCKS, TPB, 0, stream>>>(c2, partials, out);
}